// GraphEvolution_46282567582227
// MI455X (gfx1250) — compile-verified
//
#include <hip/hip_runtime.h>
#include <math.h>

typedef __attribute__((ext_vector_type(2))) float v2f;
typedef __attribute__((ext_vector_type(8))) float v8f;

#define OC 64   // out_ch per head in every GAT layer

// ---------- ordered-float <-> uint for atomic max ----------
static __device__ __forceinline__ unsigned f2ord(float f) {
    unsigned u = __float_as_uint(f);
    return (u & 0x80000000u) ? ~u : (u | 0x80000000u);
}
static __device__ __forceinline__ float ord2f(unsigned u) {
    return (u & 0x80000000u) ? __uint_as_float(u ^ 0x80000000u)
                             : __uint_as_float(~u);
}

// =====================================================================
// WMMA fp32 GEMM:  Y[n x O] = X[n x K] @ W[O x K]^T + bias
// one wave per 16(M) x 64(N) tile; K-loop register-blocked in chunks of
// 32 K-values (8 wmma steps): 40 independent b64 loads issued up front,
// then 32 back-to-back v_wmma ops, so waits amortize instead of
// serializing each wmma behind a full memory round-trip.
// Requires K%4==0, n%16==0, O%64==0.
// =====================================================================
__global__ __launch_bounds__(32)
void k_gemm_wmma(const float* __restrict__ X, const float* __restrict__ W,
                 const float* __restrict__ B, float* __restrict__ Y,
                 int K, int O)
{
    const int lane = threadIdx.x;
    const int half = lane >> 4;          // 0: K{0,1}, 1: K{2,3}
    const int l16  = lane & 15;
    const int colb = blockIdx.y * 64;

    const float* xrow = X + (size_t)(blockIdx.x * 16 + l16) * K + 2 * half;
    const float* w0 = W + (size_t)(colb +  0 + l16) * K + 2 * half;
    const float* w1 = W + (size_t)(colb + 16 + l16) * K + 2 * half;
    const float* w2 = W + (size_t)(colb + 32 + l16) * K + 2 * half;
    const float* w3 = W + (size_t)(colb + 48 + l16) * K + 2 * half;

    v8f acc0 = {}, acc1 = {}, acc2 = {}, acc3 = {};

    int kc = 0;
    for (; kc + 32 <= K; kc += 32) {
        v2f a[8], b0[8], b1[8], b2[8], b3[8];
#pragma unroll
        for (int j = 0; j < 8; ++j) {
            const int k0 = kc + 4 * j;
            a[j]  = *(const v2f*)(xrow + k0);
            b0[j] = *(const v2f*)(w0 + k0);
            b1[j] = *(const v2f*)(w1 + k0);
            b2[j] = *(const v2f*)(w2 + k0);
            b3[j] = *(const v2f*)(w3 + k0);
        }
#pragma unroll
        for (int j = 0; j < 8; ++j) {
            acc0 = __builtin_amdgcn_wmma_f32_16x16x4_f32(false, a[j], false, b0[j], (short)0, acc0, false, false);
            acc1 = __builtin_amdgcn_wmma_f32_16x16x4_f32(false, a[j], false, b1[j], (short)0, acc1, false, false);
            acc2 = __builtin_amdgcn_wmma_f32_16x16x4_f32(false, a[j], false, b2[j], (short)0, acc2, false, false);
            acc3 = __builtin_amdgcn_wmma_f32_16x16x4_f32(false, a[j], false, b3[j], (short)0, acc3, false, false);
        }
    }
    for (; kc < K; kc += 4) {            // tail (handles K=4 layer-0 GEMMs)
        v2f a  = *(const v2f*)(xrow + kc);
        v2f b0v = *(const v2f*)(w0 + kc);
        v2f b1v = *(const v2f*)(w1 + kc);
        v2f b2v = *(const v2f*)(w2 + kc);
        v2f b3v = *(const v2f*)(w3 + kc);
        acc0 = __builtin_amdgcn_wmma_f32_16x16x4_f32(false, a, false, b0v, (short)0, acc0, false, false);
        acc1 = __builtin_amdgcn_wmma_f32_16x16x4_f32(false, a, false, b1v, (short)0, acc1, false, false);
        acc2 = __builtin_amdgcn_wmma_f32_16x16x4_f32(false, a, false, b2v, (short)0, acc2, false, false);
        acc3 = __builtin_amdgcn_wmma_f32_16x16x4_f32(false, a, false, b3v, (short)0, acc3, false, false);
    }

    // C/D layout: VGPR r -> M = r + 8*half, N = l16 (+16*t)
    const float bb0 = B ? B[colb +  0 + l16] : 0.f;
    const float bb1 = B ? B[colb + 16 + l16] : 0.f;
    const float bb2 = B ? B[colb + 32 + l16] : 0.f;
    const float bb3 = B ? B[colb + 48 + l16] : 0.f;
#pragma unroll
    for (int r = 0; r < 8; ++r) {
        const int m = blockIdx.x * 16 + r + 8 * half;
        float* yr = Y + (size_t)m * O + colb + l16;
        yr[ 0] = acc0[r] + bb0;
        yr[16] = acc1[r] + bb1;
        yr[32] = acc2[r] + bb2;
        yr[48] = acc3[r] + bb3;
    }
}

// =====================================================================
// edge feature build: ea[e] = [edge_attr(2), params(4), x2[src](4), x2[dst](4)]
// =====================================================================
__global__ void k_build_ea(const float* __restrict__ edge_attr,
                           const float* __restrict__ params,
                           const float* __restrict__ x2,
                           const int* __restrict__ src, const int* __restrict__ dst,
                           float* __restrict__ ea, int E)
{
    int e = blockIdx.x * blockDim.x + threadIdx.x;
    if (e >= E) return;
    float* o = ea + (size_t)e * 14;
    o[0] = edge_attr[2 * e + 0];
    o[1] = edge_attr[2 * e + 1];
    o[2] = params[0]; o[3] = params[1]; o[4] = params[2]; o[5] = params[3];
    const float* xs = x2 + (size_t)src[e] * 4;
    const float* xd = x2 + (size_t)dst[e] * 4;
    o[6] = xs[0]; o[7] = xs[1]; o[8] = xs[2]; o[9] = xs[3];
    o[10] = xd[0]; o[11] = xd[1]; o[12] = xd[2]; o[13] = xd[3];
}

// column sums of ea (E x 14) -> efmean (finalized by k_mean_fin)
__global__ void k_ea_colsum(const float* __restrict__ ea, float* __restrict__ efmean, int E)
{
    float acc[14];
#pragma unroll
    for (int j = 0; j < 14; ++j) acc[j] = 0.f;
    for (int r = blockIdx.x * blockDim.x + threadIdx.x; r < E; r += gridDim.x * blockDim.x) {
        const float* row = ea + (size_t)r * 14;
#pragma unroll
        for (int j = 0; j < 14; ++j) acc[j] += row[j];
    }
#pragma unroll
    for (int j = 0; j < 14; ++j)
        for (int m = 16; m > 0; m >>= 1) acc[j] += __shfl_xor(acc[j], m, 32);
    if ((threadIdx.x & 31) == 0)
#pragma unroll
        for (int j = 0; j < 14; ++j) atomicAdd(&efmean[j], acc[j]);
}

__global__ void k_mean_fin(float* efmean, int E)
{
    int j = threadIdx.x;
    if (j < 14) efmean[j] *= (1.0f / (float)E);
}

// =====================================================================
// GAT per-edge score: a[e,h] = sum_c leaky(m,0.2)*att ;  amax via atomic max
// m = xl[s] + xr[d] (+ we . ef_row   for layer 0)
// =====================================================================
__global__ void k_gat_scores(const float* __restrict__ xl, const float* __restrict__ xr,
                             const float* __restrict__ att,
                             const int* __restrict__ src, const int* __restrict__ dst,
                             const float* __restrict__ ea, const float* __restrict__ efmean,
                             const float* __restrict__ we,
                             float* __restrict__ a_out, unsigned* __restrict__ amax_bits,
                             int E, int Etot, int heads)
{
    long tid = (long)blockIdx.x * blockDim.x + threadIdx.x;
    if (tid >= (long)Etot * heads) return;
    const int e = (int)(tid / heads);
    const int h = (int)(tid % heads);
    int s, d;
    if (e < E) { s = src[e]; d = dst[e]; } else { s = e - E; d = e - E; }
    const int H = heads * OC;
    const float* pl = xl + (size_t)s * H + h * OC;
    const float* pr = xr + (size_t)d * H + h * OC;

    float f[14];
    if (we) {
        const float* er = (e < E) ? (ea + (size_t)e * 14) : efmean;
#pragma unroll
        for (int j = 0; j < 14; ++j) f[j] = er[j];
    }

    float acc = 0.f;
    for (int c = 0; c < OC; ++c) {
        float m = pl[c] + pr[c];
        if (we) {
            const float* wr = we + (size_t)(h * OC + c) * 14;
            float s2 = 0.f;
#pragma unroll
            for (int j = 0; j < 14; ++j) s2 = fmaf(f[j], wr[j], s2);
            m += s2;
        }
        const float lr = m > 0.f ? m : 0.2f * m;
        acc = fmaf(lr, att[h * OC + c], acc);
    }
    a_out[tid] = acc;
    atomicMax(&amax_bits[(size_t)d * heads + h], f2ord(acc));
}

__global__ void k_gat_exp(const float* __restrict__ a_in, const unsigned* __restrict__ amax_bits,
                          const int* __restrict__ dst,
                          float* __restrict__ ae, float* __restrict__ denom,
                          int E, int Etot, int heads)
{
    long tid = (long)blockIdx.x * blockDim.x + threadIdx.x;
    if (tid >= (long)Etot * heads) return;
    const int e = (int)(tid / heads);
    const int h = (int)(tid % heads);
    const int d = (e < E) ? dst[e] : (e - E);
    const float v = expf(a_in[tid] - ord2f(amax_bits[(size_t)d * heads + h]));
    ae[tid] = v;
    atomicAdd(&denom[(size_t)d * heads + h], v);
}

__global__ void k_gat_agg(const float* __restrict__ xl,
                          const float* __restrict__ ae, const float* __restrict__ denom,
                          const int* __restrict__ src, const int* __restrict__ dst,
                          float* __restrict__ out, int E, int Etot, int heads)
{
    const int H = heads * OC;
    long tid = (long)blockIdx.x * blockDim.x + threadIdx.x;
    if (tid >= (long)Etot * H) return;
    const int e = (int)(tid / H);
    const int c = (int)(tid % H);
    const int h = c >> 6;
    int s, d;
    if (e < E) { s = src[e]; d = dst[e]; } else { s = e - E; d = e - E; }
    const float alpha = ae[(size_t)e * heads + h] /
                        (denom[(size_t)d * heads + h] + 1e-16f);
    atomicAdd(&out[(size_t)d * H + c], xl[(size_t)s * H + c] * alpha);
}

// =====================================================================
// small elementwise / init kernels
// =====================================================================
__global__ void k_fill_u32(unsigned* p, unsigned v, long cnt)
{ long i = (long)blockIdx.x * blockDim.x + threadIdx.x; if (i < cnt) p[i] = v; }

__global__ void k_fill_f32(float* p, float v, long cnt)
{ long i = (long)blockIdx.x * blockDim.x + threadIdx.x; if (i < cnt) p[i] = v; }

__global__ void k_rowbias(float* __restrict__ out, const float* __restrict__ bias, int H, long cnt)
{ long i = (long)blockIdx.x * blockDim.x + threadIdx.x; if (i < cnt) out[i] = bias[i % H]; }

__global__ void k_elu(float* p, long cnt)
{ long i = (long)blockIdx.x * blockDim.x + threadIdx.x;
  if (i < cnt) { float v = p[i]; p[i] = v > 0.f ? v : (expf(v) - 1.f); } }

__global__ void k_relu(float* p, long cnt)
{ long i = (long)blockIdx.x * blockDim.x + threadIdx.x;
  if (i < cnt) { float v = p[i]; p[i] = v > 0.f ? v : 0.f; } }

__global__ void k_leaky(float* p, float slope, long cnt)
{ long i = (long)blockIdx.x * blockDim.x + threadIdx.x;
  if (i < cnt) { float v = p[i]; p[i] = v > 0.f ? v : slope * v; } }

__global__ void k_tanh(float* p, long cnt)
{ long i = (long)blockIdx.x * blockDim.x + threadIdx.x;
  if (i < cnt) p[i] = tanhf(p[i]); }

// mem = ydec = y2 + z  (z = [x2, 0...])
__global__ void k_memz(const float* __restrict__ y2, const float* __restrict__ x2,
                       float* __restrict__ mem, float* __restrict__ ydec, long cnt)
{
    long i = (long)blockIdx.x * blockDim.x + threadIdx.x;
    if (i >= cnt) return;
    const int c = (int)(i & 63);
    const long node = i >> 6;
    float v = y2[i] + (c < 4 ? x2[node * 4 + c] : 0.f);
    mem[i] = v; ydec[i] = v;
}

__global__ void k_addz(float* __restrict__ t, const float* __restrict__ x2, long cnt)
{
    long i = (long)blockIdx.x * blockDim.x + threadIdx.x;
    if (i >= cnt) return;
    const int c = (int)(i & 63);
    if (c < 4) t[i] += x2[(i >> 6) * 4 + c];
}

// =====================================================================
// Y = LayerNorm(X + T) ; rows of 64, one wave32 per row, 2 channels/lane
// =====================================================================
__global__ __launch_bounds__(128)
void k_add_ln(const float* __restrict__ X, const float* __restrict__ T,
              const float* __restrict__ g, const float* __restrict__ b,
              float* __restrict__ Y, int n)
{
    const int lane = threadIdx.x & 31;
    const int wid  = threadIdx.x >> 5;
    const int row  = blockIdx.x * 4 + wid;
    if (row >= n) return;
    const size_t base = (size_t)row * 64;
    const int c0 = lane * 2;
    float v0 = X[base + c0]     + T[base + c0];
    float v1 = X[base + c0 + 1] + T[base + c0 + 1];
    float s = v0 + v1;
    for (int m = 16; m > 0; m >>= 1) s += __shfl_xor(s, m, 32);
    const float mean = s * (1.0f / 64.0f);
    const float d0 = v0 - mean, d1 = v1 - mean;
    float vs = d0 * d0 + d1 * d1;
    for (int m = 16; m > 0; m >>= 1) vs += __shfl_xor(vs, m, 32);
    const float inv = rsqrtf(vs * (1.0f / 64.0f) + 1e-5f);
    Y[base + c0]     = d0 * inv * g[c0]     + b[c0];
    Y[base + c0 + 1] = d1 * inv * g[c0 + 1] + b[c0 + 1];
}

// final head: out = t @ r2_w^T + r2_b (64->4), wrap channels 0,1 into [0,1]
__global__ void k_head(const float* __restrict__ t, const float* __restrict__ w,
                       const float* __restrict__ b, float* __restrict__ out, int n)
{
    long i = (long)blockIdx.x * blockDim.x + threadIdx.x;
    if (i >= (long)n * 4) return;
    const int c = (int)(i & 3);
    const long node = i >> 2;
    const float* tr = t + node * 64;
    const float* wr = w + (size_t)c * 64;
    float acc = b[c];
#pragma unroll 8
    for (int k = 0; k < 64; ++k) acc = fmaf(tr[k], wr[k], acc);
    if (c < 2) {
        if (acc < 0.f) acc += 1.f;
        if (acc > 1.f) acc -= 1.f;
    }
    out[i] = acc;
}

// =====================================================================
// host side
// =====================================================================
static inline float* carve(char*& p, size_t nfloats)
{
    float* r = (float*)p;
    p += ((nfloats * sizeof(float) + 255) / 256) * 256;
    return r;
}
static inline int ceil_div_l(long a, long b) { return (int)((a + b - 1) / b); }

extern "C" void kernel_launch(void* const* d_in, const int* in_sizes, int n_in,
                              void* d_out, int out_size, void* d_ws, size_t ws_size,
                              hipStream_t stream)
{
    int idx = 0;
    const float* x2        = (const float*)d_in[idx++];      // (1,N,4) flat
    const int*   edge_idx  = (const int*)  d_in[idx++];      // (2,E)
    const float* edge_attr = (const float*)d_in[idx++];      // (E,2)
    const float* params    = (const float*)d_in[idx++];      // (4,)

    const int N = in_sizes[0] / 4;
    const int E = in_sizes[1] / 2;
    const int Etot = E + N;
    const int* src = edge_idx;
    const int* dst = edge_idx + E;

    struct GatP { const float *wl, *bl, *wr, *br, *att, *bias, *we; };
    GatP gat[3];
    for (int i = 0; i < 3; ++i) {
        gat[i].wl   = (const float*)d_in[idx++];
        gat[i].bl   = (const float*)d_in[idx++];
        gat[i].wr   = (const float*)d_in[idx++];
        gat[i].br   = (const float*)d_in[idx++];
        gat[i].att  = (const float*)d_in[idx++];
        gat[i].bias = (const float*)d_in[idx++];
        gat[i].we   = (i == 0) ? (const float*)d_in[idx++] : nullptr;
    }
    struct MhaP { const float *in_w, *in_b, *out_w, *out_b; };
    struct DecP { MhaP sa, ca;
                  const float *ln1_g, *ln1_b, *ln2_g, *ln2_b, *ln3_g, *ln3_b;
                  const float *ff1_w, *ff1_b, *ff2_w, *ff2_b; };
    DecP dec[2];
    for (int i = 0; i < 2; ++i) {
        dec[i].sa.in_w  = (const float*)d_in[idx++];
        dec[i].sa.in_b  = (const float*)d_in[idx++];
        dec[i].sa.out_w = (const float*)d_in[idx++];
        dec[i].sa.out_b = (const float*)d_in[idx++];
        dec[i].ca.in_w  = (const float*)d_in[idx++];
        dec[i].ca.in_b  = (const float*)d_in[idx++];
        dec[i].ca.out_w = (const float*)d_in[idx++];
        dec[i].ca.out_b = (const float*)d_in[idx++];
        dec[i].ln1_g = (const float*)d_in[idx++];
        dec[i].ln1_b = (const float*)d_in[idx++];
        dec[i].ln2_g = (const float*)d_in[idx++];
        dec[i].ln2_b = (const float*)d_in[idx++];
        dec[i].ln3_g = (const float*)d_in[idx++];
        dec[i].ln3_b = (const float*)d_in[idx++];
        dec[i].ff1_w = (const float*)d_in[idx++];
        dec[i].ff1_b = (const float*)d_in[idx++];
        dec[i].ff2_w = (const float*)d_in[idx++];
        dec[i].ff2_b = (const float*)d_in[idx++];
    }
    const float* r1_w  = (const float*)d_in[idx++];
    const float* r1_b  = (const float*)d_in[idx++];
    const float* r11_w = (const float*)d_in[idx++];
    const float* r11_b = (const float*)d_in[idx++];
    const float* r2_w  = (const float*)d_in[idx++];
    const float* r2_b  = (const float*)d_in[idx++];

    // ---- workspace carve ----
    char* wp = (char*)d_ws;
    float* xl     = carve(wp, (size_t)N * 512);
    float* xr     = carve(wp, (size_t)N * 512);
    float* bufA   = carve(wp, (size_t)N * 512);
    float* bufB   = carve(wp, (size_t)N * 512);
    float* ea     = carve(wp, (size_t)E * 14);
    float* efmean = carve(wp, 64);
    float* abuf   = carve(wp, (size_t)Etot * 8);
    float* aebuf  = carve(wp, (size_t)Etot * 8);
    unsigned* amax = (unsigned*)carve(wp, (size_t)N * 8);
    float* denom  = carve(wp, (size_t)N * 8);
    float* memb   = carve(wp, (size_t)N * 64);
    float* ydec   = carve(wp, (size_t)N * 64);
    float* t1     = carve(wp, (size_t)N * 64);
    float* t2     = carve(wp, (size_t)N * 64);

    const int BT = 256;

    // ---- edge features + column mean ----
    k_build_ea<<<ceil_div_l(E, BT), BT, 0, stream>>>(edge_attr, params, x2, src, dst, ea, E);
    k_fill_f32<<<1, 64, 0, stream>>>(efmean, 0.f, 14);
    k_ea_colsum<<<120, BT, 0, stream>>>(ea, efmean, E);
    k_mean_fin<<<1, 32, 0, stream>>>(efmean, E);

    // ---- 3 GATv2 layers ----
    const float* y_in = x2;
    float* y_outs[3] = { bufA, bufB, bufA };
    const int ics[3]   = { 4, 512, 512 };
    const int headsL[3] = { 8, 8, 1 };
    for (int L = 0; L < 3; ++L) {
        const int heads = headsL[L];
        const int H = heads * OC;
        const int K = ics[L];
        float* y_out = y_outs[L];
        dim3 gg(N / 16, H / 64);
        k_gemm_wmma<<<gg, 32, 0, stream>>>(y_in, gat[L].wl, gat[L].bl, xl, K, H);
        k_gemm_wmma<<<gg, 32, 0, stream>>>(y_in, gat[L].wr, gat[L].br, xr, K, H);
        k_fill_u32<<<ceil_div_l((long)N * heads, BT), BT, 0, stream>>>(amax, 0u, (long)N * heads);
        k_fill_f32<<<ceil_div_l((long)N * heads, BT), BT, 0, stream>>>(denom, 0.f, (long)N * heads);
        k_gat_scores<<<ceil_div_l((long)Etot * heads, BT), BT, 0, stream>>>(
            xl, xr, gat[L].att, src, dst, ea, efmean, gat[L].we,
            abuf, amax, E, Etot, heads);
        k_gat_exp<<<ceil_div_l((long)Etot * heads, BT), BT, 0, stream>>>(
            abuf, amax, dst, aebuf, denom, E, Etot, heads);
        k_rowbias<<<ceil_div_l((long)N * H, BT), BT, 0, stream>>>(y_out, gat[L].bias, H, (long)N * H);
        k_gat_agg<<<ceil_div_l((long)Etot * H, BT), BT, 0, stream>>>(
            xl, aebuf, denom, src, dst, y_out, E, Etot, heads);
        k_elu<<<ceil_div_l((long)N * H, BT), BT, 0, stream>>>(y_out, (long)N * H);
        y_in = y_out;
    }
    float* y2 = y_outs[2];          // (N,64), elu applied

    // ---- mem = ydec = y2 + z ----
    const long n64 = (long)N * 64;
    k_memz<<<ceil_div_l(n64, BT), BT, 0, stream>>>(y2, x2, memb, ydec, n64);

    // ---- decoder layers (seq len 1 => attn == out(v(.)) ) ----
    dim3 gd(N / 16, 1);
    const int lnb = ceil_div_l(N, 4);
    for (int L = 0; L < 2; ++L) {
        const DecP& p = dec[L];
        // self "attention"
        k_gemm_wmma<<<gd, 32, 0, stream>>>(ydec, p.sa.in_w + 2 * 64 * 64, p.sa.in_b + 128, t1, 64, 64);
        k_gemm_wmma<<<gd, 32, 0, stream>>>(t1, p.sa.out_w, p.sa.out_b, t2, 64, 64);
        k_add_ln<<<lnb, 128, 0, stream>>>(ydec, t2, p.ln1_g, p.ln1_b, ydec, N);
        // cross "attention" on mem
        k_gemm_wmma<<<gd, 32, 0, stream>>>(memb, p.ca.in_w + 2 * 64 * 64, p.ca.in_b + 128, t1, 64, 64);
        k_gemm_wmma<<<gd, 32, 0, stream>>>(t1, p.ca.out_w, p.ca.out_b, t2, 64, 64);
        k_add_ln<<<lnb, 128, 0, stream>>>(ydec, t2, p.ln2_g, p.ln2_b, ydec, N);
        // feed-forward
        k_gemm_wmma<<<gd, 32, 0, stream>>>(ydec, p.ff1_w, p.ff1_b, t1, 64, 64);
        k_relu<<<ceil_div_l(n64, BT), BT, 0, stream>>>(t1, n64);
        k_gemm_wmma<<<gd, 32, 0, stream>>>(t1, p.ff2_w, p.ff2_b, t2, 64, 64);
        k_add_ln<<<lnb, 128, 0, stream>>>(ydec, t2, p.ln3_g, p.ln3_b, ydec, N);
    }

    // ---- readout head ----
    k_tanh<<<ceil_div_l(n64, BT), BT, 0, stream>>>(ydec, n64);
    k_gemm_wmma<<<gd, 32, 0, stream>>>(ydec, r1_w, r1_b, t1, 64, 64);
    k_addz<<<ceil_div_l(n64, BT), BT, 0, stream>>>(t1, x2, n64);
    k_leaky<<<ceil_div_l(n64, BT), BT, 0, stream>>>(t1, 0.01f, n64);
    k_gemm_wmma<<<gd, 32, 0, stream>>>(t1, r11_w, r11_b, t2, 64, 64);
    k_leaky<<<ceil_div_l(n64, BT), BT, 0, stream>>>(t2, 0.01f, n64);
    k_head<<<ceil_div_l((long)N * 4, BT), BT, 0, stream>>>(t2, r2_w, r2_b, (float*)d_out, N);
}